// STBlock_noSatt_82867099009464
// MI455X (gfx1250) — compile-verified
//
#include <hip/hip_runtime.h>
#include <hip/hip_bf16.h>

typedef float v2f __attribute__((ext_vector_type(2)));
typedef float v8f __attribute__((ext_vector_type(8)));

#define NB 8192      // nodes
#define BT 128       // B*T1 = 8*16 columns of the GEMM B matrix
#define KC 32        // K-chunk staged in LDS per iteration
#define KSPLIT 4     // K-dimension split across blockIdx.y
#define KSEG (NB / KSPLIT)          // 2048
#define NBT (NB * BT)               // 1048576 elements per [N,BT] matrix

// ---------------- kernel 1: repack X[b,n,1,t] -> xr[n, b*16+t] ----------------
__global__ __launch_bounds__(256) void repack_x_kernel(const float* __restrict__ X,
                                                       float* __restrict__ xr) {
  int idx = blockIdx.x * 256 + threadIdx.x;   // [0, 8192*128)
  int n = idx >> 7;
  int c = idx & 127;
  int b = c >> 4;
  int t = c & 15;
  xr[idx] = X[(size_t)b * (NB * 16) + (size_t)n * 16 + t];
}

// ---------------- kernel 2: deg / dis / Adiag ----------------
__global__ __launch_bounds__(256) void degree_kernel(const float* __restrict__ A,
                                                     float* __restrict__ dis,
                                                     float* __restrict__ Adiag) {
  int wave = threadIdx.x >> 5;
  int lane = threadIdx.x & 31;
  int row = blockIdx.x * 8 + wave;
  const float* Arow = A + (size_t)row * NB;
  float s = 0.f;
  for (int j = lane * 4; j < NB; j += 32 * 4) {
    float4 v = *reinterpret_cast<const float4*>(Arow + j);
    s += v.x + v.y + v.z + v.w;
  }
  for (int off = 16; off > 0; off >>= 1) s += __shfl_xor(s, off, 32);
  if (lane == 0) {
    float diag = Arow[row];
    float deg = s - diag;               // A0 = A with zero diagonal
    Adiag[row] = diag;
    dis[row] = (deg > 0.f) ? rsqrtf(deg) : 0.f;
  }
}

// ---------------- kernels 3a/4a: K-split partial GEMM --------
// Part[seg] = A[:, seg] @ (dis * Bsrc)[seg, :]   (raw partial sums, no epilogue)
__global__ __launch_bounds__(256)
void lhat_gemm_kernel(const float* __restrict__ A,
                      const float* __restrict__ Bsrc,
                      const float* __restrict__ dis,
                      float* __restrict__ Part) {
  // pair-interleaved B panel: ldsB[k/2][n][k&1], so (B[k][n],B[k+1][n]) is one b64
  __shared__ float ldsB[KC / 2][BT][2];   // 16 KB

  const int tid  = threadIdx.x;
  const int wave = tid >> 5;
  const int lane = tid & 31;
  const int half = lane >> 4;             // 0: lanes 0-15, 1: lanes 16-31
  const int l16  = lane & 15;
  const int rowTile = blockIdx.x * 128 + wave * 16;
  const int kOff = blockIdx.y * KSEG;     // K segment for this block
  float* PartSeg = Part + (size_t)blockIdx.y * NBT;

  v8f acc[8];
  #pragma unroll
  for (int i = 0; i < 8; ++i)
    #pragma unroll
    for (int j = 0; j < 8; ++j) acc[i][j] = 0.f;

  // panel loader mapping: thread -> (k row, 16-col chunk)
  const int lk  = tid >> 3;               // 0..31
  const int ln0 = (tid & 7) * 16;         // 0..112

  for (int kb = 0; kb < KSEG; kb += KC) {
    // ---- stage dis-scaled B panel into LDS ----
    {
      const int krow = kOff + kb + lk;
      float dk = dis[krow];
      const float4* src4 =
          reinterpret_cast<const float4*>(Bsrc + (size_t)krow * BT + ln0);
      int p = lk >> 1;
      int jj = lk & 1;
      #pragma unroll
      for (int u = 0; u < 4; ++u) {
        float4 v = src4[u];
        ldsB[p][ln0 + u * 4 + 0][jj] = v.x * dk;
        ldsB[p][ln0 + u * 4 + 1][jj] = v.y * dk;
        ldsB[p][ln0 + u * 4 + 2][jj] = v.z * dk;
        ldsB[p][ln0 + u * 4 + 3][jj] = v.w * dk;
      }
    }
    __syncthreads();

    // ---- A operands: 8 x float2 per lane (16x4 WMMA A layout) ----
    const float* Abase = A + (size_t)(rowTile + l16) * NB + kOff + kb + half * 2;
    if (kb + KC < KSEG) __builtin_prefetch(Abase + KC, 0, 0);  // global_prefetch_b8
    v2f a[8];
    #pragma unroll
    for (int s = 0; s < 8; ++s)
      a[s] = *reinterpret_cast<const v2f*>(Abase + s * 4);

    // ---- 8 K-steps x 8 column tiles; B operands double-buffered in regs ----
    v2f bq[2][8];
    #pragma unroll
    for (int n0 = 0; n0 < 8; ++n0)
      bq[0][n0] = *reinterpret_cast<const v2f*>(&ldsB[half][n0 * 16 + l16][0]);

    #pragma unroll
    for (int s = 0; s < 8; ++s) {
      const int cur = s & 1;
      const int nxt = cur ^ 1;
      if (s < 7) {                         // prefetch next step's B operands
        const int p = (s + 1) * 2 + half;
        #pragma unroll
        for (int n0 = 0; n0 < 8; ++n0)
          bq[nxt][n0] = *reinterpret_cast<const v2f*>(&ldsB[p][n0 * 16 + l16][0]);
      }
      #pragma unroll
      for (int n0 = 0; n0 < 8; ++n0)
        acc[n0] = __builtin_amdgcn_wmma_f32_16x16x4_f32(
            false, a[s], false, bq[cur][n0], (short)0, acc[n0], false, false);
    }
    __syncthreads();
  }

  // ---- write raw partial sums ----
  const int r0 = rowTile + half * 8;
  #pragma unroll
  for (int g = 0; g < 8; ++g) {
    const int r = r0 + g;
    #pragma unroll
    for (int n0 = 0; n0 < 8; ++n0) {
      const int c = n0 * 16 + l16;
      PartSeg[(size_t)r * BT + c] = acc[n0][g];
    }
  }
}

// ---------------- kernels 3b/4b: reduce K-split partials + Lhat epilogue --------
// acc = sum_seg Part[seg];  D = alpha*dis[r]*(acc - Adiag[r]*dis[r]*Bsrc[r,c]) + beta*Xadd
__global__ __launch_bounds__(256)
void ksplit_epilogue_kernel(const float* __restrict__ Part,
                            const float* __restrict__ Bsrc,
                            const float* __restrict__ dis,
                            const float* __restrict__ Adiag,
                            const float* __restrict__ Xadd,
                            float* __restrict__ D,
                            float alpha, float beta) {
  int idx = blockIdx.x * 256 + threadIdx.x;   // [0, NBT)
  int r = idx >> 7;
  float acc = Part[idx] + Part[idx + NBT] + Part[idx + 2 * NBT] + Part[idx + 3 * NBT];
  float dr = dis[r];
  float ad = Adiag[r];
  float v = acc - ad * dr * Bsrc[idx];
  D[idx] = alpha * dr * v + beta * Xadd[idx];
}

// ---------------- kernel 5: Cheb mix (WMMA) + bias + relu + conv1d(K=3) + relu ----
__global__ __launch_bounds__(256)
void cheb_conv_kernel(const float* __restrict__ Tx0, const float* __restrict__ Tx1,
                      const float* __restrict__ Tx2,
                      const float* __restrict__ Wc, const float* __restrict__ bc,
                      const float* __restrict__ wt, const float* __restrict__ bt,
                      float* __restrict__ out) {
  __shared__ float sW[3 * 256];
  __shared__ float sbc[16];
  __shared__ float swt[4];
  __shared__ float stile[8][16][17];     // per-wave 16x16 tile, padded

  const int tid = threadIdx.x;
  for (int i = tid; i < 768; i += 256) sW[i] = Wc[i];
  if (tid < 16) sbc[tid] = bc[tid];
  if (tid < 3) swt[tid] = wt[tid];
  if (tid == 3) swt[3] = bt[0];
  __syncthreads();

  const int wave = tid >> 5;
  const int lane = tid & 31;
  const int half = lane >> 4;
  const int l16  = lane & 15;
  const int tileId = blockIdx.x * 8 + wave;   // 0..4095
  const int b  = tileId >> 9;                 // batch
  const int nT = tileId & 511;                // 16-row node tile

  const float* Ts[3] = {Tx0, Tx1, Tx2};
  v8f acc;
  #pragma unroll
  for (int j = 0; j < 8; ++j) acc[j] = 0.f;

  #pragma unroll
  for (int m = 0; m < 3; ++m) {
    const float* Abase =
        Ts[m] + (size_t)(nT * 16 + l16) * BT + b * 16 + half * 2;
    #pragma unroll
    for (int s = 0; s < 4; ++s) {
      v2f a = *reinterpret_cast<const v2f*>(Abase + s * 4);
      const int k = s * 4 + half * 2;
      v2f w;
      w[0] = sW[m * 256 + k * 16 + l16];
      w[1] = sW[m * 256 + (k + 1) * 16 + l16];
      acc = __builtin_amdgcn_wmma_f32_16x16x4_f32(
          false, a, false, w, (short)0, acc, false, false);
    }
  }

  // bias + relu, stage to per-wave LDS tile
  const float bcol = sbc[l16];
  #pragma unroll
  for (int g = 0; g < 8; ++g) {
    float v = acc[g] + bcol;
    stile[wave][g + half * 8][l16] = v > 0.f ? v : 0.f;
  }
  __syncthreads();

  // temporal conv K=3, pad=1, then relu
  const float w0 = swt[0], w1 = swt[1], w2 = swt[2], bb = swt[3];
  #pragma unroll
  for (int g = 0; g < 8; ++g) {
    const int r = g + half * 8;
    const int t = l16;
    float v = bb + w1 * stile[wave][r][t];
    if (t > 0)  v += w0 * stile[wave][r][t - 1];
    if (t < 15) v += w2 * stile[wave][r][t + 1];
    v = v > 0.f ? v : 0.f;
    const int n = nT * 16 + r;
    out[((size_t)b * NB + n) * 16 + t] = v;
  }
}

extern "C" void kernel_launch(void* const* d_in, const int* in_sizes, int n_in,
                              void* d_out, int out_size, void* d_ws, size_t ws_size,
                              hipStream_t stream) {
  const float* X  = (const float*)d_in[0];   // [8,8192,1,16]
  const float* A  = (const float*)d_in[1];   // [8192,8192]
  const float* Wc = (const float*)d_in[2];   // [3,16,16]
  const float* bc = (const float*)d_in[3];   // [16]
  const float* wt = (const float*)d_in[4];   // [3]
  const float* bt = (const float*)d_in[5];   // [1]
  float* out = (float*)d_out;

  float* ws    = (float*)d_ws;
  float* xr    = ws;                         // NBT
  float* Tx1   = ws + (size_t)NBT;           // NBT
  float* Tx2   = ws + (size_t)2 * NBT;       // NBT
  float* dis   = ws + (size_t)3 * NBT;       // 8192
  float* Adiag = dis + NB;                   // 8192
  float* Part  = Adiag + NB;                 // KSPLIT * NBT

  repack_x_kernel<<<4096, 256, 0, stream>>>(X, xr);
  degree_kernel<<<1024, 256, 0, stream>>>(A, dis, Adiag);

  // Tx1 = Lhat @ x
  lhat_gemm_kernel<<<dim3(64, KSPLIT), 256, 0, stream>>>(A, xr, dis, Part);
  ksplit_epilogue_kernel<<<4096, 256, 0, stream>>>(Part, xr, dis, Adiag, xr, Tx1,
                                                   -1.f, 0.f);
  // Tx2 = 2 * Lhat @ Tx1 - Tx0
  lhat_gemm_kernel<<<dim3(64, KSPLIT), 256, 0, stream>>>(A, Tx1, dis, Part);
  ksplit_epilogue_kernel<<<4096, 256, 0, stream>>>(Part, Tx1, dis, Adiag, xr, Tx2,
                                                   -2.f, -1.f);

  cheb_conv_kernel<<<512, 256, 0, stream>>>(xr, Tx1, Tx2, Wc, bc, wt, bt, out);
}